// FarthestPointSampling_38001870635075
// MI455X (gfx1250) — compile-verified
//
#include <hip/hip_runtime.h>
#include <stdint.h>

// Farthest Point Sampling, persistent cooperative kernel for gfx1250.
// - mindist resident in VGPRs (128 per thread) for all 1024 iterations
// - points streamed per-iteration through triple-buffered LDS via
//   global_load_async_to_lds_b128 (ASYNCcnt); row offsets carried in the
//   instruction IOFFSET (applies to both LDS and global sides) so only two
//   address VGPRs are live per chunk -> no VGPR-MSB pressure
// - per-thread argmax as (bestm, bestk); 64-bit packed key built once per
//   iteration for wave32 shuffle reduce + cross-WG global_atomic_max_u64

#define BATCHES   32
#define NPTS      262144
#define NSAMP     1024
#define NWG       8                       // workgroups per batch
#define BLOCK     256                     // 8 wave32
#define PT        (NPTS / (NWG * BLOCK))  // 128 points per thread
#define NCH       8                       // chunks per iteration
#define CPT       (PT / NCH)              // 16 points / thread / chunk
#define CHUNK_PTS   (BLOCK * CPT)         // 4096 points
#define CHUNK_BYTES (CHUNK_PTS * 12)      // 49152 B
#define NBUF      3                       // triple buffer -> 1 barrier per chunk
#define SMEM_BYTES  (NBUF * CHUNK_BYTES + 128)

// One async b128 row: 256 threads x 16B = 4KB. OFF is a literal byte offset
// added by hardware to BOTH the LDS destination and the global address.
#define ASYNC_ROW(lv, gv, sb, OFF)                                   \
  asm volatile("global_load_async_to_lds_b128 %0, %1, %2 offset:" #OFF \
               :: "v"(lv), "v"(gv), "s"(sb) : "memory")

__device__ __forceinline__ void issue_chunk(unsigned l, unsigned g,
                                            unsigned long long sb) {
  ASYNC_ROW(l, g, sb, 0);
  ASYNC_ROW(l, g, sb, 4096);
  ASYNC_ROW(l, g, sb, 8192);
  ASYNC_ROW(l, g, sb, 12288);
  ASYNC_ROW(l, g, sb, 16384);
  ASYNC_ROW(l, g, sb, 20480);
  ASYNC_ROW(l, g, sb, 24576);
  ASYNC_ROW(l, g, sb, 28672);
  ASYNC_ROW(l, g, sb, 32768);
  ASYNC_ROW(l, g, sb, 36864);
  ASYNC_ROW(l, g, sb, 40960);
  ASYNC_ROW(l, g, sb, 45056);   // 12 rows = 48KB chunk
}

__device__ __forceinline__ unsigned long long wave_max_u64(unsigned long long v) {
  #pragma unroll
  for (int m = 16; m >= 1; m >>= 1) {
    unsigned lo = __shfl_xor((unsigned)(v & 0xFFFFFFFFu), m, 32);
    unsigned hi = __shfl_xor((unsigned)(v >> 32), m, 32);
    unsigned long long o = ((unsigned long long)hi << 32) | lo;
    if (o > v) v = o;
  }
  return v;
}

__global__ __launch_bounds__(BLOCK) void fps_persistent_kernel(
    const float* __restrict__ points,          // (B, P, 3)
    float* __restrict__ outIdx,                // (B, NSAMP) as float
    float* __restrict__ outXYZ,                // (B, NSAMP, 3)
    unsigned long long* __restrict__ slot,     // (B, NSAMP) packed argmax
    unsigned* __restrict__ cnt)                // (B, NSAMP) arrival counters
{
  extern __shared__ __align__(16) unsigned char smemraw[];
  float* buf = (float*)smemraw;                                   // 3 x 48KB
  unsigned long long* red = (unsigned long long*)(smemraw + NBUF * CHUNK_BYTES);
  unsigned long long* bc  = red + 8;

  const int t  = threadIdx.x;
  const int b  = blockIdx.x >> 3;          // batch
  const int wg = blockIdx.x & (NWG - 1);   // workgroup within batch
  const unsigned wgbase = (unsigned)wg * (unsigned)(NPTS / NWG);  // 32768

  // Global base of this WG's contiguous 384KB point slab.
  const unsigned long long slab =
      (unsigned long long)(uintptr_t)(points + ((size_t)b * NPTS + wgbase) * 3);
  // Dynamic-LDS byte offset (== static LDS size) for raw async-to-LDS writes.
  const unsigned lds0 = (unsigned)__builtin_amdgcn_groupstaticsize();
  const unsigned t16  = (unsigned)t * 16u;

  // Per-chunk base addresses: LDS cycles over 3 buffers, global walks 8 chunks.
  auto issue = [&](int c) {
    issue_chunk(lds0 + (unsigned)(c % NBUF) * CHUNK_BYTES + t16,
                (unsigned)c * CHUNK_BYTES + t16, slab);
  };

  // Register-resident running min squared distance, one slab slice per thread.
  float md[PT];
  #pragma unroll
  for (int k = 0; k < PT; k++) md[k] = 1e10f;

  const unsigned invbase = 0xFFFFFFFFu - (wgbase + (unsigned)t);  // ~batch-local idx
  unsigned cur = 0u;                                              // first pick is 0

  for (int s = 0; s < NSAMP; s++) {
    // Centroid = points[b, cur] (uniform, served from L2).
    const float* cp = points + ((size_t)b * NPTS + cur) * 3;
    float cx = cp[0], cy = cp[1], cz = cp[2];

    if (wg == 0 && t == 0) {  // emit current pick + its coords
      outIdx[(size_t)b * NSAMP + s] = (float)cur;
      float* o = outXYZ + ((size_t)b * NSAMP + s) * 3;
      o[0] = cx; o[1] = cy; o[2] = cz;
    }

    float bestm = -1.0f;   // all distances >= 0
    int   bestk = 0;

    issue(0);
    issue(1);
    #pragma unroll
    for (int c = 0; c < NCH; c++) {
      // Chunks complete in issue order per wave: <=12 outstanding => chunk c done.
      if (c < NCH - 1) asm volatile("s_wait_asynccnt 12" ::: "memory");
      else             asm volatile("s_wait_asynccnt 0"  ::: "memory");
      __syncthreads();   // chunk c visible to all waves; and everyone is done
                         // reading buffer (c+2)%3 (it held chunk c-1).
      if (c + 2 < NCH) issue(c + 2);
      const float* cb = buf + (size_t)(c % NBUF) * (CHUNK_PTS * 3);
      #pragma unroll
      for (int j = 0; j < CPT; j++) {
        const int k = c * CPT + j;                 // md register index
        const int e = (j * BLOCK + t) * 3;         // LDS float offset
        float dx = cb[e]     - cx;
        float dy = cb[e + 1] - cy;
        float dz = cb[e + 2] - cz;
        float d  = dx * dx + dy * dy + dz * dz;
        float m  = md[k];
        m = d < m ? d : m;
        md[k] = m;
        // Strict > keeps the earliest k: matches argmax first-index tie-break.
        bool gt = m > bestm;
        bestm = gt ? m : bestm;
        bestk = gt ? k : bestk;
      }
    }

    // Build the packed key once: dist bits high, ~global_idx low.
    unsigned long long best =
        ((unsigned long long)__float_as_uint(bestm) << 32)
      | (unsigned long long)(invbase - (unsigned)(bestk * BLOCK));

    // Wave32 shuffle reduce, then cross-wave via LDS, then cross-WG via L2 atomic.
    best = wave_max_u64(best);
    if ((t & 31) == 0) red[t >> 5] = best;
    __syncthreads();
    if (t == 0) {
      unsigned long long v = red[0];
      #pragma unroll
      for (int i = 1; i < (BLOCK / 32); i++) if (red[i] > v) v = red[i];
      unsigned long long* sp = slot + (size_t)b * NSAMP + s;
      unsigned*           kp = cnt  + (size_t)b * NSAMP + s;
      atomicMax(sp, v);
      __hip_atomic_fetch_add(kp, 1u, __ATOMIC_ACQ_REL, __HIP_MEMORY_SCOPE_AGENT);
      while (__hip_atomic_load(kp, __ATOMIC_ACQUIRE, __HIP_MEMORY_SCOPE_AGENT) <
             (unsigned)NWG)
        __builtin_amdgcn_s_sleep(2);
      *bc = __hip_atomic_load(sp, __ATOMIC_RELAXED, __HIP_MEMORY_SCOPE_AGENT);
    }
    __syncthreads();   // bc visible; also fences buffer reuse by next iteration
    cur = 0xFFFFFFFFu - (unsigned)((*bc) & 0xFFFFFFFFull);  // unpack winner idx
  }
}

extern "C" void kernel_launch(void* const* d_in, const int* in_sizes, int n_in,
                              void* d_out, int out_size, void* d_ws, size_t ws_size,
                              hipStream_t stream) {
  (void)in_sizes; (void)n_in; (void)out_size; (void)ws_size;
  const float* points = (const float*)d_in[0];

  float* outIdx = (float*)d_out;                       // (32,1024) idx as float
  float* outXYZ = outIdx + (size_t)BATCHES * NSAMP;    // (32,1024,3)

  unsigned long long* slot = (unsigned long long*)d_ws;                   // 256 KB
  unsigned* cnt = (unsigned*)((char*)d_ws + (size_t)BATCHES * NSAMP * 8); // 128 KB

  // Zero sync scratch every call (capture-safe memset node).
  hipMemsetAsync(d_ws, 0, (size_t)BATCHES * NSAMP * 12, stream);

  hipFuncSetAttribute(reinterpret_cast<const void*>(fps_persistent_kernel),
                      hipFuncAttributeMaxDynamicSharedMemorySize, SMEM_BYTES);

  fps_persistent_kernel<<<dim3(BATCHES * NWG), dim3(BLOCK), SMEM_BYTES, stream>>>(
      points, outIdx, outXYZ, slot, cnt);
}